// CameraExtrinsics_61495341744219
// MI455X (gfx1250) — compile-verified
//
#include <hip/hip_runtime.h>
#include <cstdint>

// ---------------------------------------------------------------------------
// CameraExtrinsics on MI455X (gfx1250)
//
// Memory-bound problem: ~100 B of HBM traffic per point vs ~40 FLOPs.
// Strategy:
//   Kernel A: build a 1000-entry Rodrigues table (R + t, 12 floats = 48 B per
//             camera) once in d_ws  -> removes per-point sin/cos entirely.
//   Kernel B: each workgroup stages the 48 KB table into LDS with CDNA5
//             global_load_async_to_lds_b128 (+ s_wait_asynccnt), then streams
//             points: idx gather from LDS via ds_load_b128, two 3x3 matvecs,
//             coalesced stores to the 4 output regions.
// WMMA intentionally unused: every point has a unique 3x3 matrix, so there is
// no shared GEMM operand; the kernel is HBM-limited regardless.
// ---------------------------------------------------------------------------

#define TAB_ENTRIES 1024              // 1000 cameras, padded for exact tiling
#define TAB_F4      (TAB_ENTRIES * 3) // 3 float4 per entry
#define TAB_BYTES   (TAB_ENTRIES * 48)

// Entry layout (3 x float4):
//   q0 = {R00, R01, R02, R10}
//   q1 = {R11, R12, R20, R21}
//   q2 = {R22, tx,  ty,  tz }

__global__ __launch_bounds__(256) void rodrigues_table_kernel(
    const float* __restrict__ params, int nparams, float4* __restrict__ tab) {
  int e = blockIdx.x * blockDim.x + threadIdx.x;
  if (e >= nparams) return;
  const float* p = params + e * 6;
  float a = p[0], b = p[1], c = p[2];
  float tx = p[3], ty = p[4], tz = p[5];

  // K = [[0,a,b],[-a,0,c],[-b,-c,0]],  R = I + sin_c*K + cos_c*K*K
  float t2 = a * a + b * b + c * c;
  float t = sqrtf(t2);
  bool small = t < 1e-8f;
  float st = small ? 1.0f : t;
  float sn = sinf(st);
  float cs = cosf(st);
  float sin_c = small ? 1.0f : sn / st;
  float cos_c = small ? 0.5f : (1.0f - cs) / (st * st);

  float aa = a * a, bb = b * b, cc = c * c;
  float ab = a * b, ac = a * c, bc = b * c;

  float R00 = 1.0f - cos_c * (aa + bb);
  float R01 = sin_c * a - cos_c * bc;
  float R02 = sin_c * b + cos_c * ac;
  float R10 = -sin_c * a - cos_c * bc;
  float R11 = 1.0f - cos_c * (aa + cc);
  float R12 = sin_c * c - cos_c * ab;
  float R20 = -sin_c * b + cos_c * ac;
  float R21 = -sin_c * c - cos_c * ab;
  float R22 = 1.0f - cos_c * (bb + cc);

  tab[e * 3 + 0] = make_float4(R00, R01, R02, R10);
  tab[e * 3 + 1] = make_float4(R11, R12, R20, R21);
  tab[e * 3 + 2] = make_float4(R22, tx, ty, tz);
}

__global__ __launch_bounds__(256) void transform_kernel(
    const float4* __restrict__ tab, const int* __restrict__ idx,
    const float* __restrict__ o, const float* __restrict__ dvec,
    float* __restrict__ out, int N) {
  __shared__ float4 s_tab[TAB_F4];  // 48 KB -> 6 workgroups per WGP (320 KB)

  // --- CDNA5 async global->LDS copy of the whole table (3072 x 16B chunks,
  // --- 256 lanes x 12 iterations), tracked by ASYNCcnt.
  {
    unsigned lds_base = (unsigned)(uintptr_t)(&s_tab[0]);  // low 32 bits = LDS offset
    unsigned long long gbase = (unsigned long long)tab;
#pragma unroll
    for (int k = 0; k < (TAB_BYTES / 16) / 256; ++k) {  // 12 iters
      unsigned off = (threadIdx.x + (unsigned)k * 256u) * 16u;
      unsigned lds_addr = lds_base + off;
      // GVS mode: mem = SADDR(64) + VADDR(32); dest LDS byte address in VGPR.
      asm volatile("global_load_async_to_lds_b128 %0, %1, %2"
                   :
                   : "v"(lds_addr), "v"(off), "s"(gbase)
                   : "memory");
    }
    asm volatile("s_wait_asynccnt 0" ::: "memory");
  }
  __syncthreads();

  float* __restrict__ out_o = out;                    // new_o : 3N
  float* __restrict__ out_d = out + 3 * (size_t)N;    // new_d : 3N
  float* __restrict__ out_t = out + 6 * (size_t)N;    // trans : 3N
  float* __restrict__ out_R = out + 9 * (size_t)N;    // R     : 9N

  int base = blockIdx.x * 1024 + threadIdx.x;  // 4 points per thread
#pragma unroll
  for (int k = 0; k < 4; ++k) {
    int n = base + k * 256;
    if (n < N) {
      int ix = idx[n];
      // 48 B gather from LDS: 3 x ds_load_b128
      float4 q0 = s_tab[ix * 3 + 0];
      float4 q1 = s_tab[ix * 3 + 1];
      float4 q2 = s_tab[ix * 3 + 2];

      float ox = o[3 * n + 0], oy = o[3 * n + 1], oz = o[3 * n + 2];
      float dx = dvec[3 * n + 0], dy = dvec[3 * n + 1], dz = dvec[3 * n + 2];

      float R00 = q0.x, R01 = q0.y, R02 = q0.z, R10 = q0.w;
      float R11 = q1.x, R12 = q1.y, R20 = q1.z, R21 = q1.w;
      float R22 = q2.x, tx = q2.y, ty = q2.z, tz = q2.w;

      float nox = fmaf(R00, ox, fmaf(R01, oy, fmaf(R02, oz, tx)));
      float noy = fmaf(R10, ox, fmaf(R11, oy, fmaf(R12, oz, ty)));
      float noz = fmaf(R20, ox, fmaf(R21, oy, fmaf(R22, oz, tz)));

      float ndx = fmaf(R00, dx, fmaf(R01, dy, R02 * dz));
      float ndy = fmaf(R10, dx, fmaf(R11, dy, R12 * dz));
      float ndz = fmaf(R20, dx, fmaf(R21, dy, R22 * dz));

      out_o[3 * n + 0] = nox;
      out_o[3 * n + 1] = noy;
      out_o[3 * n + 2] = noz;

      out_d[3 * n + 0] = ndx;
      out_d[3 * n + 1] = ndy;
      out_d[3 * n + 2] = ndz;

      out_t[3 * n + 0] = tx;
      out_t[3 * n + 1] = ty;
      out_t[3 * n + 2] = tz;

      out_R[9 * n + 0] = R00;
      out_R[9 * n + 1] = R01;
      out_R[9 * n + 2] = R02;
      out_R[9 * n + 3] = R10;
      out_R[9 * n + 4] = R11;
      out_R[9 * n + 5] = R12;
      out_R[9 * n + 6] = R20;
      out_R[9 * n + 7] = R21;
      out_R[9 * n + 8] = R22;
    }
  }
}

extern "C" void kernel_launch(void* const* d_in, const int* in_sizes, int n_in,
                              void* d_out, int out_size, void* d_ws, size_t ws_size,
                              hipStream_t stream) {
  const float* params = (const float*)d_in[0];  // (size, 6) f32
  const int* idx = (const int*)d_in[1];         // (N,) int
  const float* o = (const float*)d_in[2];       // (N, 3) f32
  const float* dvec = (const float*)d_in[3];    // (N, 3) f32

  int nparams = in_sizes[0] / 6;  // 1000
  int N = in_sizes[1];            // 4194304

  // Rodrigues table lives at the start of d_ws (needs TAB_BYTES = 48 KB).
  float4* tab = (float4*)d_ws;
  (void)ws_size;

  int blkA = (nparams + 255) / 256;
  rodrigues_table_kernel<<<blkA, 256, 0, stream>>>(params, nparams, tab);

  int blkB = (N + 1023) / 1024;  // 4 points per thread, 256 threads per block
  transform_kernel<<<blkB, 256, 0, stream>>>(tab, idx, o, dvec, (float*)d_out, N);
}